// GCNClassifier_48902497632371
// MI455X (gfx1250) — compile-verified
//
#include <hip/hip_runtime.h>
#include <hip/hip_bf16.h>

typedef float v2f __attribute__((ext_vector_type(2)));
typedef float v8f __attribute__((ext_vector_type(8)));

#define IN_F   128
#define F1     64
#define F2     128
#define F3     256
#define EMB    256
#define NGRAPH 64

// ---------------- small elementwise kernels ----------------

__global__ void k_fill(float* __restrict__ p, float v, int n) {
  int i = blockIdx.x * blockDim.x + threadIdx.x;
  if (i < n) p[i] = v;
}

__global__ void k_count_deg(const int* __restrict__ dst, float* __restrict__ deg, int E) {
  int e = blockIdx.x * blockDim.x + threadIdx.x;
  if (e < E) atomicAdd(&deg[dst[e]], 1.0f);
}

__global__ void k_rsqrt_inplace(float* __restrict__ d, int n) {
  int i = blockIdx.x * blockDim.x + threadIdx.x;
  if (i < n) d[i] = rsqrtf(d[i]);   // deg >= 1 always (self loops)
}

__global__ void k_edge_norm(const int* __restrict__ src, const int* __restrict__ dst,
                            const float* __restrict__ dinv, float* __restrict__ nrm, int E) {
  int e = blockIdx.x * blockDim.x + threadIdx.x;
  if (e < E) nrm[e] = dinv[src[e]] * dinv[dst[e]];
}

// ---------------- fp32 WMMA GEMM: Out[M,FO] = X[M,FI] @ W[FI,FO] (+ bias) ----------------
// Block = 4 waves = 64 rows x 64 cols. W slab staged in LDS pre-swizzled into
// B-fragment order: lds[kstep][tile][lane*2 + j] so each lane issues one
// conflict-free ds_load_b64 per (kstep,tile).
//
// V_WMMA_F32_16X16X4_F32 layouts (ISA 7.12.2):
//   A 16x4 : lanes 0-15 row M=lane, K={0,1}; lanes 16-31 K={2,3}
//   B 4x16 : lanes 0-15 col N=lane, K={0,1}; lanes 16-31 K={2,3}
//   C/D    : VGPR r -> M=r (lanes 0-15) / r+8 (lanes 16-31), N=lane&15
template <int FI, int FO>
__global__ __launch_bounds__(128) void k_wmma_gemm(
    const float* __restrict__ X, const float* __restrict__ W,
    const float* __restrict__ bias, float* __restrict__ Out, int M) {
  constexpr int KSTEPS = FI / 4;
  __shared__ float lds[FI * 64];        // <= 64 KB (FI=256)

  const int wave = threadIdx.x >> 5;
  const int lane = threadIdx.x & 31;
  const int col0 = blockIdx.y * 64;

  // ---- stage + swizzle W[:, col0:col0+64] into fragment order ----
  // read coalesced row-major; write lds[((s*4+t)*64) + 2*l + j]
  for (int idx = threadIdx.x; idx < FI * 64; idx += 128) {
    int kk = idx >> 6;                  // W row
    int cc = idx & 63;                  // col within slab
    int s   = kk >> 2;
    int lhi = (kk >> 1) & 1;
    int j   = kk & 1;
    int t   = cc >> 4;
    int l   = (lhi << 4) | (cc & 15);
    lds[(((s << 2) + t) << 6) + (l << 1) + j] = W[(size_t)kk * FO + col0 + cc];
  }
  __syncthreads();

  // clamp instead of early-return: every wave passes the barrier and keeps
  // EXEC all-ones through the WMMAs; only the store is guarded (wave-uniform).
  int rowTile = blockIdx.x * 4 + wave;
  const bool active = (rowTile * 16) < M;
  if (!active) rowTile = 0;
  const int row0 = rowTile * 16;
  const int r  = lane & 15;             // A: row in tile; B/C: col in tile
  const int kb = (lane >> 4) << 1;      // half-wave K sub-block: 0 or 2

  const float* __restrict__ xrow = X + (size_t)(row0 + r) * FI + kb;

  v8f acc[4] = {};
#pragma unroll 4
  for (int s = 0; s < KSTEPS; ++s) {
    v2f a = *(const v2f*)(xrow + (s << 2));          // global_load_b64, 8B aligned
#pragma unroll
    for (int t = 0; t < 4; ++t) {
      v2f b = *(const v2f*)&lds[(((s << 2) + t) << 6) + (lane << 1)];  // ds_load_b64
      acc[t] = __builtin_amdgcn_wmma_f32_16x16x4_f32(
          false, a, false, b, (short)0, acc[t], false, false);
    }
  }

  if (active) {
    const int rbase = row0 + ((lane >> 4) << 3);     // +8 rows for upper half-wave
#pragma unroll
    for (int t = 0; t < 4; ++t) {
      const int c = col0 + t * 16 + r;
      const float bb = bias ? bias[c] : 0.0f;
#pragma unroll
      for (int rr = 0; rr < 8; ++rr) {
        Out[(size_t)(rbase + rr) * FO + c] = acc[t][rr] + bb;
      }
    }
  }
}

// ---------------- GCN aggregation (float4-vectorized, L2-resident) ----------------

// A[i,:] = G[i,:] * dinv[i]^2   (self-loop term, also initializes A)
__global__ void k_agg_self4(const float4* __restrict__ G, const float* __restrict__ dinv,
                            float4* __restrict__ A, int total4, int s2) {
  int i = blockIdx.x * blockDim.x + threadIdx.x;
  if (i < total4) {
    float d = dinv[i >> s2];
    float d2 = d * d;
    float4 g = G[i];
    A[i] = make_float4(g.x * d2, g.y * d2, g.z * d2, g.w * d2);
  }
}

// A[dst,:] += G[src,:] * norm[e]
__global__ void k_agg_edges4(const float4* __restrict__ G, const int* __restrict__ src,
                             const int* __restrict__ dst, const float* __restrict__ nrm,
                             float* __restrict__ A, long total4, int s2, int mask2) {
  long i = (long)blockIdx.x * blockDim.x + threadIdx.x;
  if (i < total4) {
    int e = (int)(i >> s2);
    int v = (int)(i & mask2);
    int sN = src[e], dN = dst[e];
    float nr = nrm[e];
    float4 g = G[(((size_t)sN) << s2) + v];
    float* a = A + ((((size_t)dN) << s2) + v) * 4;
    atomicAdd(a + 0, g.x * nr);
    atomicAdd(a + 1, g.y * nr);
    atomicAdd(a + 2, g.z * nr);
    atomicAdd(a + 3, g.w * nr);
  }
}

__global__ void k_bias_relu4(float4* __restrict__ A, const float4* __restrict__ b,
                             int total4, int mask2) {
  int i = blockIdx.x * blockDim.x + threadIdx.x;
  if (i < total4) {
    float4 a = A[i];
    float4 bb = b[i & mask2];
    A[i] = make_float4(fmaxf(a.x + bb.x, 0.0f), fmaxf(a.y + bb.y, 0.0f),
                       fmaxf(a.z + bb.z, 0.0f), fmaxf(a.w + bb.w, 0.0f));
  }
}

// segment_max per graph; post-ReLU values >= 0, so unsigned bit-pattern
// atomicMax into a zero buffer == segment_max + empty-graph guard.
__global__ void k_pool_max4(const float4* __restrict__ A, const int* __restrict__ batch,
                            float* __restrict__ pooled, int total4) {
  int i = blockIdx.x * blockDim.x + threadIdx.x;
  if (i < total4) {
    int node = i >> 6;                  // EMB/4 == 64 vecs per node
    int v = i & 63;
    int g = batch[node];
    float4 a = A[i];
    unsigned int* p = (unsigned int*)(pooled + ((size_t)(g << 6) + v) * 4);
    atomicMax(p + 0, __float_as_uint(a.x));
    atomicMax(p + 1, __float_as_uint(a.y));
    atomicMax(p + 2, __float_as_uint(a.z));
    atomicMax(p + 3, __float_as_uint(a.w));
  }
}

// out[g,c] = embd[g,:] . Wc[:,c] + bc[c]    (64x2)
__global__ void k_classifier(const float* __restrict__ embd, const float* __restrict__ Wc,
                             const float* __restrict__ bc, float* __restrict__ out) {
  int i = threadIdx.x;
  if (i < NGRAPH * 2) {
    int g = i >> 1, c = i & 1;
    const float* eg = embd + (size_t)g * EMB;
    float s = bc[c];
#pragma unroll 8
    for (int k = 0; k < EMB; ++k) s += eg[k] * Wc[k * 2 + c];
    out[i] = s;
  }
}

// ---------------- host-side orchestration ----------------

extern "C" void kernel_launch(void* const* d_in, const int* in_sizes, int n_in,
                              void* d_out, int out_size, void* d_ws, size_t ws_size,
                              hipStream_t stream) {
  const float* x     = (const float*)d_in[0];
  const int*   edge  = (const int*)d_in[1];     // [2, E] flat
  const int*   batch = (const int*)d_in[2];
  const float* W1 = (const float*)d_in[3];  const float* b1 = (const float*)d_in[4];
  const float* W2 = (const float*)d_in[5];  const float* b2 = (const float*)d_in[6];
  const float* W3 = (const float*)d_in[7];  const float* b3 = (const float*)d_in[8];
  const float* Wh = (const float*)d_in[9];  const float* bh = (const float*)d_in[10];
  const float* Wc = (const float*)d_in[11]; const float* bc = (const float*)d_in[12];

  const int N = in_sizes[0] / IN_F;   // 100000
  const int E = in_sizes[1] / 2;      // 1600000
  const int* src = edge;
  const int* dst = edge + E;

  // workspace carve-up (floats)
  float* ws     = (float*)d_ws;
  float* dinv   = ws;                          // N      (degree, then rsqrt)
  float* nrm    = dinv + N;                    // E
  float* G      = nrm + E;                     // N*256  (GEMM output)
  float* A      = G + (size_t)N * 256;         // N*256  (aggregated activation)
  float* pooled = A + (size_t)N * 256;         // 64*256

  float* out_logits = (float*)d_out;           // [64,2]
  float* embd       = out_logits + NGRAPH * 2; // [64,256]

  const int T = 256;
  auto cdiv = [](long a, long b) { return (unsigned)((a + b - 1) / b); };

  // --- symmetric normalization ---
  k_fill<<<cdiv(N, T), T, 0, stream>>>(dinv, 1.0f, N);              // self loop
  k_count_deg<<<cdiv(E, T), T, 0, stream>>>(dst, dinv, E);
  k_rsqrt_inplace<<<cdiv(N, T), T, 0, stream>>>(dinv, N);
  k_edge_norm<<<cdiv(E, T), T, 0, stream>>>(src, dst, dinv, nrm, E);

  const int rowTiles = N / 16;                 // 6250 (exact)
  const unsigned gx  = cdiv(rowTiles, 4);

  // --- layer 1: 128 -> 64 ---
  k_wmma_gemm<IN_F, F1><<<dim3(gx, F1 / 64), 128, 0, stream>>>(x, W1, nullptr, G, N);
  k_agg_self4<<<cdiv((long)N * F1 / 4, T), T, 0, stream>>>((const float4*)G, dinv, (float4*)A, N * (F1 / 4), 4);
  k_agg_edges4<<<cdiv((long)E * F1 / 4, T), T, 0, stream>>>((const float4*)G, src, dst, nrm, A, (long)E * (F1 / 4), 4, 15);
  k_bias_relu4<<<cdiv((long)N * F1 / 4, T), T, 0, stream>>>((float4*)A, (const float4*)b1, N * (F1 / 4), 15);

  // --- layer 2: 64 -> 128 ---
  k_wmma_gemm<F1, F2><<<dim3(gx, F2 / 64), 128, 0, stream>>>(A, W2, nullptr, G, N);
  k_agg_self4<<<cdiv((long)N * F2 / 4, T), T, 0, stream>>>((const float4*)G, dinv, (float4*)A, N * (F2 / 4), 5);
  k_agg_edges4<<<cdiv((long)E * F2 / 4, T), T, 0, stream>>>((const float4*)G, src, dst, nrm, A, (long)E * (F2 / 4), 5, 31);
  k_bias_relu4<<<cdiv((long)N * F2 / 4, T), T, 0, stream>>>((float4*)A, (const float4*)b2, N * (F2 / 4), 31);

  // --- layer 3: 128 -> 256 ---
  k_wmma_gemm<F2, F3><<<dim3(gx, F3 / 64), 128, 0, stream>>>(A, W3, nullptr, G, N);
  k_agg_self4<<<cdiv((long)N * F3 / 4, T), T, 0, stream>>>((const float4*)G, dinv, (float4*)A, N * (F3 / 4), 6);
  k_agg_edges4<<<cdiv((long)E * F3 / 4, T), T, 0, stream>>>((const float4*)G, src, dst, nrm, A, (long)E * (F3 / 4), 6, 63);
  k_bias_relu4<<<cdiv((long)N * F3 / 4, T), T, 0, stream>>>((float4*)A, (const float4*)b3, N * (F3 / 4), 63);

  // --- global max pool ---
  k_fill<<<cdiv(NGRAPH * EMB, T), T, 0, stream>>>(pooled, 0.0f, NGRAPH * EMB);
  k_pool_max4<<<cdiv((long)N * (EMB / 4), T), T, 0, stream>>>((const float4*)A, batch, pooled, N * (EMB / 4));

  // --- head: embd = pooled @ Wh + bh  (written straight into d_out) ---
  k_wmma_gemm<EMB, EMB><<<dim3(1, EMB / 64), 128, 0, stream>>>(pooled, Wh, bh, embd, NGRAPH);

  // --- classifier: out = embd @ Wc + bc ---
  k_classifier<<<1, 128, 0, stream>>>(embd, Wc, bc, out_logits);
}